// StyleCrossAttention_807453851997
// MI455X (gfx1250) — compile-verified
//
#include <hip/hip_runtime.h>

typedef __attribute__((ext_vector_type(2))) float v2f;
typedef __attribute__((ext_vector_type(8))) float v8f;
typedef __attribute__((ext_vector_type(4))) float f4;

// Problem constants (from reference): B=16, C=512, H=W=64, ctx=256
constexpr int BATCH = 16;
constexpr int C     = 512;
constexpr int CTX   = 256;
constexpr int NPIX  = 64 * 64; // 4096

// ---------------------------------------------------------------------------
// Kernel 1: v = context @ Wkv[C:, :].T   -> (16 x 512), K = 256
// One wave (32 threads) per 16-wide column tile; 32 blocks total.
// A = context (16 x 256, row major), B[k][n] = Wkv[512 + n][k].
// f32 WMMA 16x16x4 lane layout:
//   A: lanes 0-15 -> M=lane, K={0,1}; lanes 16-31 -> M=lane-16, K={2,3}
//   B: lanes 0-15 -> N=lane, K={0,1}; lanes 16-31 -> N=lane-16, K={2,3}
//   C/D: vgpr i, lane l -> row (i + 8*(l/16)), col (l%16)
// ---------------------------------------------------------------------------
__global__ void __launch_bounds__(32)
gemm_v_kernel(const float* __restrict__ context,
              const float* __restrict__ Wkv,
              float* __restrict__ v_out) {
  const int lane = threadIdx.x;
  const int nb   = blockIdx.x * 16;   // output column tile base
  const int half = lane >> 4;         // 0 or 1
  const int lm   = lane & 15;

  const float* arow = context + (size_t)lm * CTX;              // A row m=lm
  const float* brow = Wkv + (size_t)(C + nb + lm) * CTX;       // B col n=lm

  v8f acc = {};
  for (int kb = 0; kb < CTX; kb += 4) {
    const int ks = kb + half * 2;
    v2f a, b;
    a.x = arow[ks];
    a.y = arow[ks + 1];
    b.x = brow[ks];
    b.y = brow[ks + 1];
    acc = __builtin_amdgcn_wmma_f32_16x16x4_f32(
        /*neg_a=*/false, a, /*neg_b=*/false, b,
        /*c_mod=*/(short)0, acc, /*reuse_a=*/false, /*reuse_b=*/false);
  }

#pragma unroll
  for (int i = 0; i < 8; ++i) {
    v_out[(size_t)(i + 8 * half) * C + nb + lm] = acc[i];
  }
}

// ---------------------------------------------------------------------------
// Kernel 2: y = v @ Wout.T + bout   -> (16 x 512), K = 512
// Same tiling: 32 single-wave blocks.
// ---------------------------------------------------------------------------
__global__ void __launch_bounds__(32)
gemm_y_kernel(const float* __restrict__ v,
              const float* __restrict__ Wout,
              const float* __restrict__ bout,
              float* __restrict__ y_out) {
  const int lane = threadIdx.x;
  const int nb   = blockIdx.x * 16;
  const int half = lane >> 4;
  const int lm   = lane & 15;

  const float* arow = v + (size_t)lm * C;                      // A row m=lm
  const float* brow = Wout + (size_t)(nb + lm) * C;            // B col n=lm

  v8f acc = {};
  for (int kb = 0; kb < C; kb += 4) {
    const int ks = kb + half * 2;
    v2f a, b;
    a.x = arow[ks];
    a.y = arow[ks + 1];
    b.x = brow[ks];
    b.y = brow[ks + 1];
    acc = __builtin_amdgcn_wmma_f32_16x16x4_f32(
        false, a, false, b, (short)0, acc, false, false);
  }

  const float bias = bout[nb + lm];
#pragma unroll
  for (int i = 0; i < 8; ++i) {
    y_out[(size_t)(i + 8 * half) * C + nb + lm] = acc[i] + bias;
  }
}

// ---------------------------------------------------------------------------
// Kernel 3: broadcast y[b,c] over 4096 contiguous pixels.
// out[(b*C + c)*4096 + n] = y[b*C + c].  134 MB streaming store -> use
// non-temporal float4 stores (TH=NT) to bypass cache retention.
// Grid: B*C = 8192 blocks, 256 threads; each thread stores 4 x float4.
// ---------------------------------------------------------------------------
__global__ void __launch_bounds__(256)
bcast_kernel(const float* __restrict__ y, float* __restrict__ out) {
  const int bc    = blockIdx.x;             // 0 .. 8191
  const float val = y[bc];                  // uniform -> scalarized load
  f4 v4;
  v4.x = val; v4.y = val; v4.z = val; v4.w = val;

  f4* dst = (f4*)(out + (size_t)bc * NPIX); // 4096 floats = 1024 float4
  const int t = threadIdx.x;
#pragma unroll
  for (int i = 0; i < 4; ++i) {
    __builtin_nontemporal_store(v4, dst + t + i * 256);
  }
}

// ---------------------------------------------------------------------------
// d_in order: x(unused), context, Wq(unused), Wkv, Wout, bout
// Workspace: v (16*512 f32) then y (16*512 f32) = 64 KB total.
// ---------------------------------------------------------------------------
extern "C" void kernel_launch(void* const* d_in, const int* in_sizes, int n_in,
                              void* d_out, int out_size, void* d_ws, size_t ws_size,
                              hipStream_t stream) {
  const float* context = (const float*)d_in[1];
  const float* Wkv     = (const float*)d_in[3];
  const float* Wout    = (const float*)d_in[4];
  const float* bout    = (const float*)d_in[5];
  float* out = (float*)d_out;

  float* v_buf = (float*)d_ws;                  // 16 x 512
  float* y_buf = v_buf + (size_t)BATCH * C;     // 16 x 512

  gemm_v_kernel<<<C / 16, 32, 0, stream>>>(context, Wkv, v_buf);
  gemm_y_kernel<<<C / 16, 32, 0, stream>>>(v_buf, Wout, bout, y_buf);
  bcast_kernel<<<BATCH * C, 256, 0, stream>>>(y_buf, out);
}